// VectorQuantizer_uni_2181843386754
// MI455X (gfx1250) — compile-verified
//
#include <hip/hip_runtime.h>
#include <hip/hip_bf16.h>

// Problem constants
#define TT   16
#define NN   64
#define CC   16
#define HH   32
#define WW   32
#define HW   (HH*WW)            // 1024
#define CHW  (CC*HW)            // 16384
#define NCHW (NN*CHW)           // 1048576
#define NP   (NN*HW)            // 65536 positions
#define KK   1024               // codebook size

typedef __attribute__((ext_vector_type(2))) float v2f;
typedef __attribute__((ext_vector_type(8))) float v8f;

// ws float header layout
#define F_L1SUM   0
#define F_ERRSUM  1
#define F_CHSUM   2    // 16
#define F_CHSQ    18   // 16
#define F_SCALE   34   // 16
#define F_SHIFT   50   // 16
#define F_L1FIN   66
#define F_HDR     128  // header floats

// ---------------- K0: zero accumulators ----------------
__global__ void k0_zero(float* ws) {
    int t = threadIdx.x;
    if (t < F_HDR) ws[t] = 0.0f;
}

// ---------------- K1: membrane output reduction over T ----------------
// x:(T,N,C,H,W) -> flat_x[(n*HW+hw)*16 + c]
__global__ void k1_xmem(const float* __restrict__ x,
                        const float* __restrict__ alpha_p,
                        float* __restrict__ flat_x) {
    int i = blockIdx.x * 256 + threadIdx.x;        // 0..NCHW-1
    float a  = alpha_p[0];
    float am = 1.0f - a;
    float at = a * (1.0f / (float)TT);
    float acc = 0.0f, cd = 1.0f;                   // cd = 0.8^(15-t), t descending
    for (int t = TT - 1; t >= 0; --t) {
        acc = fmaf(x[t * NCHW + i], am * cd + at, acc);
        cd *= 0.8f;
    }
    int n  = i >> 14;          // / CHW
    int c  = (i >> 10) & 15;   // channel
    int hw = i & (HW - 1);
    flat_x[((n << 10) + hw) * CC + c] = acc;
}

// ---------------- K2: VQ nearest-code via V_WMMA_F32_16X16X4_F32 ----------------
// One wave handles 16 positions vs all 1024 codes.
// A (16x16 x-tile) as four 16x4 fragments; B = emb^T tiles from LDS.
// ISA layout: A frag kb: lane<16 -> {X[m][kb],X[m][kb+1]}, lane>=16 -> {X[m][kb+2],X[m][kb+3]}
//             B frag kb: lane<16 -> {E[kb][n],E[kb+1][n]},  lane>=16 -> {E[kb+2][n],E[kb+3][n]}
//             C: VGPR r holds rows M=r (lanes 0-15) and M=r+8 (lanes 16-31), N=lane%16
__global__ void k2_vq(const float* __restrict__ flat_x,
                      const float* __restrict__ emb,
                      int* __restrict__ idx_out) {
    extern __shared__ float lds[];
    float* lds_emb = lds;            // 16384 floats (64 KB)
    float* lds_esq = lds + KK * CC;  // 1024 floats

    const int tid = threadIdx.x;     // 256 threads = 8 waves
    // stage full codebook (coalesced float4)
    {
        const float4* src = (const float4*)emb;
        float4* dst = (float4*)lds_emb;
        for (int k = tid; k < KK * CC / 4; k += 256) dst[k] = src[k];
    }
    __syncthreads();
    for (int k = tid; k < KK; k += 256) {
        float s = 0.0f;
        #pragma unroll
        for (int c = 0; c < CC; ++c) { float e = lds_emb[k * CC + c]; s = fmaf(e, e, s); }
        lds_esq[k] = s;
    }
    __syncthreads();

    const int wave = tid >> 5;
    const int lane = tid & 31;
    const int hi   = lane >> 4;     // half-wave select
    const int ln   = lane & 15;     // M for A, N for B/C
    const int p0   = (blockIdx.x * 8 + wave) * 16;   // base position of this wave's 16 rows

    // load A fragments (constant across the code loop)
    v2f a[4];
    #pragma unroll
    for (int kb = 0; kb < 4; ++kb) {
        const float* row = flat_x + (p0 + ln) * CC + kb * 4 + 2 * hi;
        a[kb].x = row[0];
        a[kb].y = row[1];
    }

    float best[8];
    int   bidx[8];
    #pragma unroll
    for (int r = 0; r < 8; ++r) { best[r] = 3.402823e38f; bidx[r] = 0; }

    for (int j = 0; j < KK / 16; ++j) {
        const int code = j * 16 + ln;
        const float* erow = lds_emb + code * CC;
        v8f acc = {};
        #pragma unroll
        for (int kb = 0; kb < 4; ++kb) {
            v2f b;
            b.x = erow[kb * 4 + 2 * hi];
            b.y = erow[kb * 4 + 2 * hi + 1];
            acc = __builtin_amdgcn_wmma_f32_16x16x4_f32(
                    false, a[kb], false, b, (short)0, acc, false, false);
        }
        const float esq = lds_esq[code];
        #pragma unroll
        for (int r = 0; r < 8; ++r) {
            float cand = fmaf(-2.0f, acc[r], esq);   // ||e||^2 - 2 x.e (||x||^2 const per row)
            if (cand < best[r]) { best[r] = cand; bidx[r] = code; }
        }
    }

    // min-reduce across the 16 lanes that share each output row
    #pragma unroll
    for (int r = 0; r < 8; ++r) {
        #pragma unroll
        for (int off = 1; off < 16; off <<= 1) {
            float ov = __shfl_xor(best[r], off, 32);
            int   oi = __shfl_xor(bidx[r], off, 32);
            if (ov < best[r] || (ov == best[r] && oi < bidx[r])) { best[r] = ov; bidx[r] = oi; }
        }
    }
    if (ln == 0) {
        #pragma unroll
        for (int r = 0; r < 8; ++r) idx_out[p0 + r + 8 * hi] = bidx[r];
    }
}

// ---------------- K3: gather quantized, commitment loss, 1x1 conv, BN partials ----------------
__global__ void k3_conv(const float* __restrict__ flat_x,
                        const int* __restrict__ idx,
                        const float* __restrict__ emb,
                        const float* __restrict__ conv_w,
                        const float* __restrict__ conv_b,
                        float* __restrict__ y_buf,   // layout n*CHW + o*HW + hw
                        float* __restrict__ ws) {
    __shared__ float s_w[CC * CC];
    __shared__ float s_b[CC];
    __shared__ float s_chsum[CC], s_chsq[CC];
    __shared__ float s_l1;
    const int tid = threadIdx.x;  // 256
    s_w[tid] = conv_w[tid];
    if (tid < CC) { s_b[tid] = conv_b[tid]; s_chsum[tid] = 0.0f; s_chsq[tid] = 0.0f; }
    if (tid == 0) s_l1 = 0.0f;
    __syncthreads();

    const int p = blockIdx.x * 256 + tid;            // 0..NP-1
    const float* er = emb + idx[p] * CC;
    const float* xr = flat_x + p * CC;
    float q[CC], l1 = 0.0f;
    #pragma unroll
    for (int c = 0; c < CC; ++c) {
        q[c] = er[c];
        float d = q[c] - xr[c];
        l1 = fmaf(d, d, l1);
    }
    const int n = p >> 10, hw = p & (HW - 1);
    #pragma unroll
    for (int o = 0; o < CC; ++o) {
        float y = s_b[o];
        #pragma unroll
        for (int c = 0; c < CC; ++c) y = fmaf(q[c], s_w[o * CC + c], y);
        y_buf[n * CHW + o * HW + hw] = y;
        atomicAdd(&s_chsum[o], y);
        atomicAdd(&s_chsq[o], y * y);
    }
    atomicAdd(&s_l1, l1);
    __syncthreads();
    if (tid < CC) {
        atomicAdd(&ws[F_CHSUM + tid], s_chsum[tid]);
        atomicAdd(&ws[F_CHSQ + tid], s_chsq[tid]);
    }
    if (tid == 0) atomicAdd(&ws[F_L1SUM], s_l1);
}

// ---------------- K4: finalize BN affine + loss1 ----------------
__global__ void k4_bn(const float* __restrict__ gamma,
                      const float* __restrict__ beta,
                      float* __restrict__ ws) {
    int o = threadIdx.x;
    if (o < CC) {
        const float cnt = (float)NP;                 // T identical copies -> same stats
        float m   = ws[F_CHSUM + o] / cnt;
        float var = ws[F_CHSQ + o] / cnt - m * m;
        float inv = rsqrtf(var + 1e-5f);
        float sc  = gamma[o] * inv;
        ws[F_SCALE + o] = sc;
        ws[F_SHIFT + o] = beta[o] - m * sc;
    }
    if (o == 0) ws[F_L1FIN] = 1.25f * ws[F_L1SUM] / (float)NCHW;
}

// ---------------- K5: fused BN-apply + LIF + dual PSP + loss + spike store ----------------
__global__ void k5_lif(const float* __restrict__ x,
                       const float* __restrict__ y_buf,
                       const float* __restrict__ ws_ro,
                       float* __restrict__ out,
                       float* __restrict__ ws) {
    const int i = blockIdx.x * 256 + threadIdx.x;    // 0..NCHW-1, same layout as y_buf
    const int o = (i >> 10) & 15;
    const float yb = fmaf(y_buf[i], ws_ro[F_SCALE + o], ws_ro[F_SHIFT + o]);
    float v = 0.0f, ss = 0.0f, sx = 0.0f, err = 0.0f;
    #pragma unroll
    for (int t = 0; t < TT; ++t) {
        if (t < TT - 1) __builtin_prefetch(x + (t + 1) * NCHW + i, 0, 0);
        v = 0.5f * (v + yb);                         // v += (yb - v)/tau, tau=2
        float s = (v >= 1.0f) ? 1.0f : 0.0f;         // H(v - V_TH)
        v = (1.0f - s) * v;                          // hard reset
        ss = 0.5f * (ss + s);                        // psp(spikes)
        sx = 0.5f * (sx + x[t * NCHW + i]);          // psp(x)
        float d = ss - sx;
        err = fmaf(d, d, err);
        out[t * NCHW + i] = s;
    }
    #pragma unroll
    for (int off = 16; off > 0; off >>= 1) err += __shfl_down(err, off, 32);
    if ((threadIdx.x & 31) == 0) atomicAdd(&ws[F_ERRSUM], err);
}

// ---------------- K6: final loss scalar ----------------
__global__ void k6_loss(float* __restrict__ out, const float* __restrict__ ws) {
    out[TT * NCHW] = ws[F_L1FIN] + 1.25f * ws[F_ERRSUM] / (float)(TT * NCHW);
}

extern "C" void kernel_launch(void* const* d_in, const int* in_sizes, int n_in,
                              void* d_out, int out_size, void* d_ws, size_t ws_size,
                              hipStream_t stream) {
    const float* x       = (const float*)d_in[0];   // (T,N,C,H,W)
    const float* alpha   = (const float*)d_in[1];   // scalar
    const float* emb     = (const float*)d_in[2];   // (K,C)
    const float* conv_w  = (const float*)d_in[3];   // (C,C)
    const float* conv_b  = (const float*)d_in[4];
    const float* gamma   = (const float*)d_in[5];
    const float* beta    = (const float*)d_in[6];
    float* out = (float*)d_out;

    float* ws     = (float*)d_ws;
    float* flat_x = ws + F_HDR;                 // NP*CC floats (4 MB)
    float* y_buf  = flat_x + NP * CC;           // NCHW floats  (4 MB)
    int*   idxbuf = (int*)(y_buf + NCHW);       // NP ints

    k0_zero<<<1, 128, 0, stream>>>(ws);
    k1_xmem<<<NCHW / 256, 256, 0, stream>>>(x, alpha, flat_x);
    // 8 waves/block * 16 rows/wave = 128 positions/block -> 512 blocks
    size_t lds_bytes = (KK * CC + KK) * sizeof(float);   // 68 KB (LDS: 320 KB/WGP)
    k2_vq<<<NP / 128, 256, lds_bytes, stream>>>(flat_x, emb, idxbuf);
    k3_conv<<<NP / 256, 256, 0, stream>>>(flat_x, idxbuf, emb, conv_w, conv_b, y_buf, ws);
    k4_bn<<<1, 32, 0, stream>>>(gamma, beta, ws);
    k5_lif<<<NCHW / 256, 256, 0, stream>>>(x, y_buf, ws, out, ws);
    k6_loss<<<1, 1, 0, stream>>>(out, ws);
}